// WeightedDILATEloss_35476429865783
// MI455X (gfx1250) — compile-verified
//
#include <hip/hip_runtime.h>

typedef __attribute__((ext_vector_type(2))) float v2f;
typedef __attribute__((ext_vector_type(8))) float v8f;

#define BIGF 1.0e9f
#define KINV 144.26950408889634f    // 1/(gamma*ln2), gamma = 0.01
#define GLN2 0.006931471805599453f  // gamma*ln2
#define NSEQ 1024
#define NT   64                      // 64x64 grid of 16x16 tiles

// Process one 16x16 DP tile: WMMA builds the weighted cost tile into LDS,
// then a 31-step wave-local anti-diagonal scan does the soft-DTW recurrence.
__device__ __forceinline__ void dtw_tile(
    int TI, int TJ, int T, int lane, float tm, float xv,
    float rowIn, float colIn, float corner,
    float* __restrict__ dws, float* __restrict__ sROW,
    float* __restrict__ sCOL, float* __restrict__ sCB,
    const float* __restrict__ sW, float* __restrict__ res)
{
  const int n15 = lane & 15;
  const int hi  = lane >> 4;

  // ---- WMMA: (t_i - x_j)^2 = [t^2, 1, t] . [1, x^2, -2x]^T  (K=4, K3 unused)
  // A 16x4 f32 layout: lanes 0-15 -> {K0,K1}, lanes 16-31 -> {K2,K3}, M = lane&15
  v2f a, bmat;
  a.x    = (hi == 0) ? tm * tm : tm;          // K0 = t^2 | K2 = t
  a.y    = (hi == 0) ? 1.0f    : 0.0f;        // K1 = 1   | K3 = 0
  bmat.x = (hi == 0) ? 1.0f    : -2.0f * xv;  // K0 = 1   | K2 = -2x
  bmat.y = (hi == 0) ? xv * xv : 0.0f;        // K1 = x^2 | K3 = 0
  v8f c = {};
  c = __builtin_amdgcn_wmma_f32_16x16x4_f32(false, a, false, bmat, (short)0, c,
                                            false, false);

  // Scale by logistic weight, park in LDS. Row stride 18 -> diagonal reads
  // hit addr = 17*ii + s, gcd(17,64)=1 -> conflict-free.
  const int gj = TJ * 16 + n15;
#pragma unroll
  for (int r = 0; r < 8; ++r) {
    int M  = r + 8 * hi;             // C/D layout: M = r + 8*(lane>>4), N = lane&15
    int gi = TI * 16 + M;
    int dd = (gi > gj) ? (gi - gj) : (gj - gi);
    dws[M * 18 + n15] = c[r] * sW[dd];
  }

  // ---- anti-diagonal scan (lane ii owns row ii; only lanes 0-15 carry cells)
  float colsh = __shfl_up(colIn, 1);
  float rprev = BIGF, rprev2 = BIGF, row_sm1 = BIGF;
  const int ii = lane;
#pragma unroll 1
  for (int s = 0; s < 31; ++s) {
    float up_sh = __shfl_up(rprev, 1);   // R(i-1, j)   from lane ii-1, step s-1
    float dg_sh = __shfl_up(rprev2, 1);  // R(i-1, j-1) from lane ii-1, step s-2
    float row_s = __shfl(rowIn, s);      // boundary row value at column s
    int  jj    = s - ii;
    bool valid = (ii < 16) && (jj >= 0) && (jj < 16);
    float left = (jj == 0) ? colIn : rprev;
    float up   = (ii == 0) ? row_s : up_sh;
    float dg   = (ii == 0) ? ((jj == 0) ? corner : row_sm1)
                           : ((jj == 0) ? colsh  : dg_sh);
    int   daddr = valid ? (ii * 18 + jj) : 0;
    float dval  = dws[daddr];
    float m    = fminf(dg, fminf(up, left));
    float sum  = __builtin_amdgcn_exp2f((m - dg) * KINV) +
                 __builtin_amdgcn_exp2f((m - up) * KINV) +
                 __builtin_amdgcn_exp2f((m - left) * KINV);
    float rnew = dval + m - GLN2 * __builtin_amdgcn_logf(sum);  // logf = log2
    if (valid) {
      if (ii == 15) sROW[TJ * 16 + jj] = rnew;             // bottom row out
      if (jj == 15) sCOL[TI * 16 + ii] = rnew;             // right col out
      if ((ii == 15) && (jj == 15)) {
        sCB[(T & 1) * NT + TJ] = rnew;                     // corner (parity buf)
        if ((TI == NT - 1) && (TJ == NT - 1)) *res = rnew; // R(N-1,N-1)
      }
      rprev2 = rprev;
      rprev  = rnew;
    }
    row_sm1 = row_s;
  }
}

__global__ __launch_bounds__(1024) void WeightedDILATE_sdtw_kernel(
    const float* __restrict__ inp, const float* __restrict__ tgt,
    float* __restrict__ res)
{
  __shared__ float sROW[NSEQ];        // horizontal tile-boundary frontier
  __shared__ float sCOL[NSEQ];        // vertical tile-boundary frontier
  __shared__ float sCB[2 * NT];       // corner values, double-buffered by parity
  __shared__ float sW[NSEQ];          // logistic weight table on |i-j|
  __shared__ float sD[32][16 * 18];   // per-wave WMMA cost tile (padded)

  const int b    = blockIdx.x;
  const int tid  = threadIdx.x;
  const int lane = tid & 31;
  const int w    = tid >> 5;          // wave id = tile-row owner
  const int n15  = lane & 15;

  // w(d) = 1/(1+exp(-G*(d - N/2))), G = 0.05
  sW[tid] = 1.0f / (1.0f + __expf(-0.05f * ((float)tid - 512.0f)));
  __syncthreads();

  const float* xg = inp + b * NSEQ;   // columns j <- input
  const float* tg = tgt + b * NSEQ;   // rows    i <- target
  const float tmA = tg[w * 16 + n15];         // row values for TI = w
  const float tmB = tg[(w + 32) * 16 + n15];  // row values for TI = w+32
  float* dws = &sD[w][0];

  for (int T = 0; T < 2 * NT - 1; ++T) {
    const int  TJa  = T - w;          // wave handles tiles (w, TJa), (w+32, TJb)
    const int  TJb  = T - w - 32;
    const bool actA = (TJa >= 0) && (TJa < NT);
    const bool actB = (TJb >= 0) && (TJb < NT);

    // ---- read all inter-tile boundary inputs BEFORE the barrier ----
    float xvA = 0.f, rowA = BIGF, colA = BIGF, corA = BIGF;
    float xvB = 0.f, rowB = BIGF, colB = BIGF, corB = BIGF;
    if (actA) {
      xvA = xg[TJa * 16 + n15];
      if (w > 0 && lane < 16)   rowA = sROW[TJa * 16 + lane];
      if (TJa > 0 && lane < 16) colA = sCOL[w * 16 + lane];
      corA = (w > 0) ? ((TJa > 0) ? sCB[(T & 1) * NT + TJa - 1] : BIGF)
                     : ((TJa == 0) ? 0.0f : BIGF);   // R(-1,-1) = 0
      if (TJa + 1 < NT) __builtin_prefetch(&xg[(TJa + 1) * 16 + n15], 0, 1);
    }
    if (actB) {
      xvB = xg[TJb * 16 + n15];
      if (lane < 16)            rowB = sROW[TJb * 16 + lane];  // TI=w+32 > 0
      if (TJb > 0 && lane < 16) colB = sCOL[(w + 32) * 16 + lane];
      corB = (TJb > 0) ? sCB[(T & 1) * NT + TJb - 1] : BIGF;
    }
    __syncthreads();   // all reads done before any tile overwrites boundaries

    if (actA) dtw_tile(w, TJa, T, lane, tmA, xvA, rowA, colA, corA,
                       dws, sROW, sCOL, sCB, sW, res + b);
    if (actB) dtw_tile(w + 32, TJb, T, lane, tmB, xvB, rowB, colB, corB,
                       dws, sROW, sCOL, sCB, sW, res + b);
    __syncthreads();   // boundary writes visible before next diagonal's reads
  }
}

// Deterministic ordered batch mean (no float atomics -> graph replays bit-stable).
__global__ void WeightedDILATE_reduce_kernel(const float* __restrict__ res,
                                             float* __restrict__ out)
{
  if (threadIdx.x == 0 && blockIdx.x == 0) {
    float s = 0.0f;
#pragma unroll
    for (int i = 0; i < 32; ++i) s += res[i];
    out[0] = s * (1.0f / 32.0f);
  }
}

extern "C" void kernel_launch(void* const* d_in, const int* in_sizes, int n_in,
                              void* d_out, int out_size, void* d_ws, size_t ws_size,
                              hipStream_t stream)
{
  const float* inp = (const float*)d_in[0];  // "input"  [32,1024,1]
  const float* tgt = (const float*)d_in[1];  // "target" [32,1024,1]
  float* ws = (float*)d_ws;                  // 32 per-batch soft-DTW values

  WeightedDILATE_sdtw_kernel<<<32, 1024, 0, stream>>>(inp, tgt, ws);
  WeightedDILATE_reduce_kernel<<<1, 32, 0, stream>>>(ws, (float*)d_out);
}